// Model_25056839204957
// MI455X (gfx1250) — compile-verified
//
#include <hip/hip_runtime.h>
#include <hip/hip_bf16.h>
#include <stdint.h>

#define NUM_GROUPS 64
#define TOPK       8
#define BLOCK_M    128
#define NUM_TOKENS 8192
#define HIDDEN     4096
#define NFLAT      (NUM_TOKENS * TOPK)      // 65536
#define ROW_BYTES  (HIDDEN * 2)             // 8192 bytes per bf16 row
#define MAX_SLOTS  73728                    // >= 65536 + 64*127
#define SORT_THREADS 128
#define PER_THREAD  (NFLAT / SORT_THREADS)  // 512

typedef int v4i __attribute__((ext_vector_type(4)));

// meta layout (int32): [0]=total_padded, [16..80)=start[e], [80..144)=count[e],
//                      [144..208)=padded[e], [208..272)=lastval[e]

#if defined(__gfx1250__) && \
    __has_builtin(__builtin_amdgcn_global_load_async_to_lds_b128) && \
    __has_builtin(__builtin_amdgcn_global_store_async_from_lds_b128)
#define USE_ASYNC_LDS 1
#else
#define USE_ASYNC_LDS 0
#endif

__device__ __forceinline__ void wait_asynccnt0() {
#if __has_builtin(__builtin_amdgcn_s_wait_asynccnt)
    __builtin_amdgcn_s_wait_asynccnt(0);
#else
    asm volatile("s_wait_asynccnt 0" ::: "memory");
#endif
}

// ---------------------------------------------------------------------------
// Kernel (emitted FIRST so the disasm snippet shows it): bandwidth-bound row
// gather. One 8192-byte bf16 row per block. CDNA5 path: async global->LDS->
// global B128 copies (ASYNCcnt-tracked); each lane stores exactly the LDS
// bytes it loaded, so only s_wait_asynccnt is needed (no workgroup barrier).
// ---------------------------------------------------------------------------
__global__ __launch_bounds__(256) void k_gather(
    const char* __restrict__ hidden, const int* __restrict__ meta,
    const int* __restrict__ sorted, char* __restrict__ out_bytes)
{
    const int total = meta[0];
    const int p = blockIdx.x;
    if (p >= total) return;
    const int tok = sorted[p] >> 3;                 // // TOPK
    const char* src = hidden + (size_t)tok * ROW_BYTES;
    char* dst = out_bytes + (size_t)p * ROW_BYTES;
    const int t = threadIdx.x;

#if USE_ASYNC_LDS
    __shared__ __align__(16) char buf[ROW_BYTES];
    const int o0 = t * 16;
    const int o1 = 4096 + t * 16;
    __builtin_amdgcn_global_load_async_to_lds_b128(
        (v4i*)(src + o0), (v4i*)(buf + o0), 0, 0);
    __builtin_amdgcn_global_load_async_to_lds_b128(
        (v4i*)(src + o1), (v4i*)(buf + o1), 0, 0);
    wait_asynccnt0();
    __builtin_amdgcn_global_store_async_from_lds_b128(
        (v4i*)(dst + o0), (v4i*)(buf + o0), 0, 0);
    __builtin_amdgcn_global_store_async_from_lds_b128(
        (v4i*)(dst + o1), (v4i*)(buf + o1), 0, 0);
    wait_asynccnt0();
#else
    const uint4* s4 = (const uint4*)src;
    uint4* d4 = (uint4*)dst;
    d4[t]       = s4[t];
    d4[t + 256] = s4[t + 256];
#endif
}

// ---------------------------------------------------------------------------
// Kernel: stable counting sort of 65536 expert ids into padded slots.
// Single workgroup, 128 threads, each owning a contiguous chunk of 512 ids
// (contiguity + in-order processing per thread => stability).
// ---------------------------------------------------------------------------
__global__ __launch_bounds__(SORT_THREADS) void k_sort(
    const int* __restrict__ topk_ids, int* __restrict__ meta,
    int* __restrict__ sorted, int* __restrict__ firstpos,
    char* __restrict__ out_bytes)
{
    __shared__ int cnt[SORT_THREADS * 65];   // stride 65 -> conflict-free columns
    __shared__ int expStart[64], expCount[64], expPad[64], lastv[64];
    __shared__ int totalS;
    const int t = threadIdx.x;

    for (int i = t; i < SORT_THREADS * 65; i += SORT_THREADS) cnt[i] = 0;
    if (t < 64) lastv[t] = 0;
    __syncthreads();

    // Phase 1: per-thread histogram over its contiguous chunk
    const int base = t * PER_THREAD;
    for (int k = 0; k < PER_THREAD; ++k) {
        int e = topk_ids[base + k] & (NUM_GROUPS - 1);
        cnt[t * 65 + e]++;
    }
    __syncthreads();

    // Phase 2: per-expert totals + padded counts
    if (t < 64) {
        int s = 0;
        for (int i = 0; i < SORT_THREADS; ++i) s += cnt[i * 65 + t];
        expCount[t] = s;
        expPad[t]   = ((s + BLOCK_M - 1) / BLOCK_M) * BLOCK_M;
    }
    __syncthreads();
    if (t == 0) {
        int run = 0;
        for (int e = 0; e < NUM_GROUPS; ++e) { expStart[e] = run; run += expPad[e]; }
        totalS  = run;
        meta[0] = run;
    }
    __syncthreads();

    // Phase 3: convert per-(thread,expert) counts into stable scatter offsets
    if (t < 64) {
        int run = expStart[t];
        for (int i = 0; i < SORT_THREADS; ++i) {
            int c = cnt[i * 65 + t];
            cnt[i * 65 + t] = run;
            run += c;
        }
        meta[16 + t]  = expStart[t];
        meta[80 + t]  = expCount[t];
        meta[144 + t] = expPad[t];
    }
    __syncthreads();

    // Phase 4: stable scatter (also build value -> slot inverse)
    for (int k = 0; k < PER_THREAD; ++k) {
        int idx = base + k;
        int e   = topk_ids[idx] & (NUM_GROUPS - 1);
        int pos = cnt[t * 65 + e]++;
        sorted[pos]   = idx;
        firstpos[idx] = pos;
        if (pos + 1 == expStart[e] + expCount[e]) lastv[e] = idx;  // segment's last value
    }
    __syncthreads();
    if (t < 64) meta[208 + t] = lastv[t];

    // Phase 5: padding slots repeat segment-last value (0 for empty expert),
    // and m_indices[p] = expert id for every padded slot.
    const int total = totalS;
    int* m_out = (int*)(out_bytes + (size_t)total * ROW_BYTES);
    for (int e = 0; e < NUM_GROUPS; ++e) {
        const int o = expStart[e], c = expCount[e], p = expPad[e];
        const int lvv = lastv[e];
        for (int j = t; j < p - c; j += SORT_THREADS) sorted[o + c + j] = lvv;
        for (int j = t; j < p;     j += SORT_THREADS) m_out[o + j] = e;
    }
}

// ---------------------------------------------------------------------------
// Kernel: inv_perm = argsort(sorted_token_ids, stable)[:NFLAT] without a
// second sort: rank(v) = v + sum over experts e with lastval[e] < v of the
// number of padding duplicates of that expert; duplicate occurrences of v
// sit contiguously at the expert's padding slots.
// ---------------------------------------------------------------------------
__global__ __launch_bounds__(256) void k_invperm(
    const int* __restrict__ meta, const int* __restrict__ firstpos,
    char* __restrict__ out_bytes)
{
    __shared__ int lv[64], padcnt[64], padstart[64];
    const int t = threadIdx.x;
    const int total = meta[0];
    if (t < 64) {
        int st = meta[16 + t], cn = meta[80 + t], pd = meta[144 + t];
        lv[t]       = meta[208 + t];
        padcnt[t]   = pd - cn;
        padstart[t] = st + cn;
    }
    __syncthreads();

    const int v = blockIdx.x * 256 + t;
    if (v >= NFLAT) return;

    int base = v;
    int myexp = -1;
    #pragma unroll
    for (int e = 0; e < NUM_GROUPS; ++e) {
        int pc = padcnt[e];
        if (pc > 0) {
            int l = lv[e];
            if (l < v)       base += pc;
            else if (l == v) myexp = e;
        }
    }

    int* inv = (int*)(out_bytes + (size_t)total * ROW_BYTES + (size_t)total * 4);
    if (base < NFLAT) inv[base] = firstpos[v];
    if (myexp >= 0) {
        const int pc = padcnt[myexp], ps = padstart[myexp];
        for (int k = 1; k <= pc; ++k) {
            int j = base + k;
            if (j < NFLAT) inv[j] = ps + (k - 1);
        }
    }
}

// ---------------------------------------------------------------------------
extern "C" void kernel_launch(void* const* d_in, const int* in_sizes, int n_in,
                              void* d_out, int out_size, void* d_ws, size_t ws_size,
                              hipStream_t stream) {
    const char* hidden = (const char*)d_in[0];      // bf16 [8192,4096]
    const int*  topk   = (const int*)d_in[1];       // int32 [8192,8]
    char* out = (char*)d_out;

    int* ws       = (int*)d_ws;
    int* meta     = ws;                             // 1024 ints reserved
    int* sorted   = ws + 1024;                      // MAX_SLOTS ints
    int* firstpos = ws + 1024 + MAX_SLOTS;          // NFLAT ints

    k_sort<<<1, SORT_THREADS, 0, stream>>>(topk, meta, sorted, firstpos, out);
    k_gather<<<MAX_SLOTS, 256, 0, stream>>>(hidden, meta, sorted, out);
    k_invperm<<<NFLAT / 256, 256, 0, stream>>>(meta, firstpos, out);
}